// LSEP_71545565217249
// MI455X (gfx1250) — compile-verified
//
#include <hip/hip_runtime.h>

typedef __attribute__((ext_vector_type(2))) float v2f;
typedef __attribute__((ext_vector_type(8))) float v8f;
typedef __attribute__((ext_vector_type(2))) int   v2i;

// Wave-level sum of 32 f32 lane values using the CDNA5 matrix unit.
// A (16x4 f32): lane L<16 supplies A[L,0]=v (K=1 slot zero); lane L>=16 supplies
// A[L-16,2]=v (K=3 slot zero). B = ones(4x16). Then D[m,n] = v_m + v_{m+16}.
// D layout: lanes 0-15 VGPR j = D[j,lane] = u_j ; lanes 16-31 VGPR j = u_{j+8}.
// So sum of own 8 D components gives u0..u7 (low half) / u8..u15 (high half);
// one xor-16 shuffle produces the full wave sum in every lane.
__device__ __forceinline__ float wave_sum_wmma(float v) {
  v2f a; a[0] = v;    a[1] = 0.0f;
  v2f b; b[0] = 1.0f; b[1] = 1.0f;
  v8f c = {};
  c = __builtin_amdgcn_wmma_f32_16x16x4_f32(
      /*neg_a=*/false, a, /*neg_b=*/false, b,
      /*c_mod=*/(short)0, c, /*reuse_a=*/false, /*reuse_b=*/false);
  float s = c[0] + c[1] + c[2] + c[3] + c[4] + c[5] + c[6] + c[7];
  s += __shfl_xor(s, 16, 32);
  return s;
}

// Block-level sum (result valid on thread 0). 256 threads = 8 waves.
__device__ __forceinline__ float block_sum(float v) {
  __shared__ float lds[8];
  float w = wave_sum_wmma(v);
  const int lane = threadIdx.x & 31;
  const int wid  = threadIdx.x >> 5;
  if (lane == 0) lds[wid] = w;
  __syncthreads();
  float t = 0.0f;
  if (threadIdx.x == 0) {
    const int nw = blockDim.x >> 5;
    for (int i = 0; i < nw; ++i) t += lds[i];
  }
  return t;
}

__global__ void lsep_main(const float* __restrict__ T,
                          const long long* __restrict__ bayes,
                          const int* __restrict__ partial,
                          float* __restrict__ blockPartials,
                          int Bn) {
  const int stride = gridDim.x * blockDim.x;
  float local = 0.0f;

  for (int i = blockIdx.x * blockDim.x + threadIdx.x; i < Bn; i += stride) {
    const int bsel = (int)bayes[i];  // 0..9
    // Row gather: T + i*100 + bsel*10 -> byte offset 400*i + 40*bsel (8B aligned)
    const v2f* __restrict__ qrow =
        reinterpret_cast<const v2f*>(T + (size_t)i * 100 + (size_t)bsel * 10);
    // partial + i*10 -> byte offset 40*i (8B aligned, coalesced across lanes)
    const v2i* __restrict__ prow =
        reinterpret_cast<const v2i*>(partial + (size_t)i * 10);

    float sneg = 0.0f, spos = 0.0f;
#pragma unroll
    for (int k = 0; k < 5; ++k) {
      v2f qq = qrow[k];
      v2i pp = prow[k];
#pragma unroll
      for (int j = 0; j < 2; ++j) {
        const float q = qq[j];
        const int   p = pp[j];            // 0 => neg set, 1 => pos set
        const float sgn = (p != 0) ? -1.0f : 1.0f;
        const float e = __expf(q * sgn);  // exp(q) or exp(-q), one transcendental
        sneg += (p == 0) ? e : 0.0f;
        spos += (p != 0) ? e : 0.0f;
      }
    }
    local += log1pf(sneg * spos);
  }

  const float bt = block_sum(local);
  if (threadIdx.x == 0) blockPartials[blockIdx.x] = bt;
}

__global__ void lsep_final(const float* __restrict__ partials, int n,
                           float* __restrict__ out, float invB) {
  float local = 0.0f;
  for (int i = threadIdx.x; i < n; i += blockDim.x) local += partials[i];
  const float t = block_sum(local);
  if (threadIdx.x == 0) *out = t * invB;
}

extern "C" void kernel_launch(void* const* d_in, const int* in_sizes, int n_in,
                              void* d_out, int out_size, void* d_ws, size_t ws_size,
                              hipStream_t stream) {
  const float*     T       = (const float*)d_in[0];
  const long long* bayes   = (const long long*)d_in[1];  // int64
  const int*       partial = (const int*)d_in[2];
  float*           out     = (float*)d_out;
  float*           pbuf    = (float*)d_ws;

  const int Bn = in_sizes[1];  // batch count from bayes

  int blocks = 2048;
  const int needed = (Bn + 255) / 256;
  if (blocks > needed) blocks = needed;
  const int wsCap = (int)(ws_size / sizeof(float));
  if (blocks > wsCap) blocks = wsCap;
  if (blocks < 1) blocks = 1;

  lsep_main<<<blocks, 256, 0, stream>>>(T, bayes, partial, pbuf, Bn);
  lsep_final<<<1, 256, 0, stream>>>(pbuf, blocks, out, 1.0f / (float)Bn);
}